// SelfAttention_34376918237561
// MI455X (gfx1250) — compile-verified
//
#include <hip/hip_runtime.h>

// ---------------------------------------------------------------------------
// SelfAttention for MI455X (gfx1250), fp32 path using V_WMMA_F32_16X16X4_F32
// + double-buffered Tensor Data Mover (tensor_load_to_lds) GEMM pipeline.
// B=2, T=1024, C=512, NH=128, HD=4  (scale = 1/HD = 0.25 per the reference)
// ---------------------------------------------------------------------------

typedef float v2f __attribute__((ext_vector_type(2)));
typedef float v8f __attribute__((ext_vector_type(8)));
typedef unsigned int v4u __attribute__((ext_vector_type(4)));
typedef int v4i __attribute__((ext_vector_type(4)));
typedef int v8i __attribute__((ext_vector_type(8)));

static constexpr int Bb = 2;
static constexpr int Tt = 1024;
static constexpr int Cc = 512;
static constexpr int NH = 128;
static constexpr int HD = 4;

__device__ __forceinline__ v8f wmma_f32_16x16x4(v2f a, v2f b, v8f c) {
  // 8 args: (neg_a, A, neg_b, B, c_mod, C, reuse_a, reuse_b)
  return __builtin_amdgcn_wmma_f32_16x16x4_f32(false, a, false, b,
                                               (short)0, c, false, false);
}

// ---------------------------------------------------------------------------
// TDM: DMA one 2D fp32 tile [nrows x 32] (row stride `row_stride` elements)
// from global memory into LDS at byte offset `lds_addr`, inserting 4 DWORDs
// of padding after every 32 DWORDs -> LDS row stride = 36 floats (144 B,
// 16B-aligned rows, conflict-free 16-lane column reads).
// D# packing per CDNA5 ISA (groups 0/1; 2D => groups 2/3
// zero/disabled). Tracked by TENSORcnt.
// ---------------------------------------------------------------------------
__device__ __forceinline__ void tdm_load_tile_f32(unsigned lds_addr,
                                                  const float* gsrc,
                                                  unsigned row_stride,
                                                  unsigned nrows) {
  const unsigned long long ga = (unsigned long long)(size_t)gsrc;
  v4u g0;
  g0.x = 1u;                                  // count=1 (valid), user mode
  g0.y = lds_addr;                            // lds_addr [63:32]
  g0.z = (unsigned)ga;                        // global_addr [95:64]
  g0.w = (unsigned)(ga >> 32) | (2u << 30);   // global_addr[56:32] | type=2
  v8i g1;
  // data_size=2 (4B) | pad_enable | pad_interval=4 (32 dw) | pad_amount=3 (4 dw)
  g1[0] = (2 << 16) | (1 << 20) | (4 << 22) | (3 << 25);
  g1[1] = (int)((row_stride & 0xffffu) << 16);                  // tensor_dim0 lo
  g1[2] = (int)(((row_stride >> 16) & 0xffffu) |
                ((nrows & 0xffffu) << 16));                     // dim0 hi | dim1 lo
  g1[3] = (int)(((nrows >> 16) & 0xffffu) | (32u << 16));       // dim1 hi | tile_dim0=32
  g1[4] = (int)(nrows & 0xffffu);                               // tile_dim1 | tile_dim2=0
  g1[5] = (int)row_stride;                                      // tensor_dim0_stride lo
  g1[6] = 0;                                                    // stride hi | dim1_stride lo
  g1[7] = 0;
  const v4i z4 = {};
#if defined(__clang_major__) && (__clang_major__ >= 23)
  const v8i z8 = {};
  __builtin_amdgcn_tensor_load_to_lds(g0, g1, z4, z4, z8, 0);
#else
  __builtin_amdgcn_tensor_load_to_lds(g0, g1, z4, z4, 0);
#endif
}

// ---------------------------------------------------------------------------
// Kernel 1/3: O[M,N] = A[M,K] * W[N,K]^T   (fp32 WMMA, TDM double-buffered)
// Block: 256 threads = 8 waves, 2(M) x 4(N); block tile 32(M) x 64(N).
// Pipeline: wave0 issues TDM for K-slice k+1 into buffer b^1, all waves run
// 8 WMMAs from buffer b, wave0 drains TENSORcnt, block barrier publishes.
// Requires M%32==0, N%64==0, K%32==0 (true for both uses here).
// ---------------------------------------------------------------------------
__global__ __launch_bounds__(256) void gemm_xwt(const float* __restrict__ A,
                                                const float* __restrict__ W,
                                                float* __restrict__ O,
                                                int M, int N, int K) {
  __shared__ float As[2][32 * 36];
  __shared__ float Ws[2][64 * 36];

  const int t    = threadIdx.x;
  const int lane = t & 31;
  const int w    = t >> 5;
  const int wm   = w >> 2;        // 0..1
  const int wn   = w & 3;         // 0..3
  const int ml   = lane & 15;     // M (A) / N (B) index within fragment
  const int g    = lane >> 4;     // K-pair group: K = g*2, g*2+1

  const long bm = (long)blockIdx.y * 32;
  const long bn = (long)blockIdx.x * 64;

  const unsigned asOff0 = (unsigned)(size_t)(&As[0][0]);
  const unsigned wsOff0 = (unsigned)(size_t)(&Ws[0][0]);
  const unsigned asStep = 32u * 36u * 4u;
  const unsigned wsStep = 64u * 36u * 4u;

  // Preload K-slice 0 into buffer 0.
  if (w == 0) {
    tdm_load_tile_f32(asOff0, A + bm * K, (unsigned)K, 32u);
    tdm_load_tile_f32(wsOff0, W + bn * K, (unsigned)K, 64u);
    __builtin_amdgcn_s_wait_tensorcnt(0);
  }
  __syncthreads();

  v8f acc = {};
  int buf = 0;

  for (int k0 = 0; k0 < K; k0 += 32) {
    const bool more = (k0 + 32) < K;
    if (w == 0) {
      if (more) {
        // Issue next slice's DMA into the other buffer; overlap with compute.
        tdm_load_tile_f32(asOff0 + (unsigned)(buf ^ 1) * asStep,
                          A + bm * K + k0 + 32, (unsigned)K, 32u);
        tdm_load_tile_f32(wsOff0 + (unsigned)(buf ^ 1) * wsStep,
                          W + bn * K + k0 + 32, (unsigned)K, 64u);
      }
    } else if (w <= 3 && (k0 + 64) < K) {
      // Waves 1-3 warm L2 for the slice after the one in flight (96 rows).
      const int idx = (w - 1) * 32 + lane;  // 0..95
      const float* prow =
          (idx < 32) ? (A + (bm + idx) * K) : (W + (bn + (idx - 32)) * K);
      __builtin_prefetch(prow + k0 + 64, 0, 1);
    }

    const float* Ac = &As[buf][0];
    const float* Wc = &Ws[buf][0];
#pragma unroll
    for (int ks = 0; ks < 8; ++ks) {
      // A frag: lane ml = row, VGPR pair holds K = (ks*4 + g*2) and +1
      v2f af = *(const v2f*)(&Ac[(wm * 16 + ml) * 36 + ks * 4 + g * 2]);
      // B frag: lane ml = col n, B[k][n] = W[n][k]
      v2f bf = *(const v2f*)(&Wc[(wn * 16 + ml) * 36 + ks * 4 + g * 2]);
      acc = wmma_f32_16x16x4(af, bf, acc);
    }

    if (w == 0 && more) __builtin_amdgcn_s_wait_tensorcnt(0);
    __syncthreads();  // publish next buffer; WAR fence for the one just read
    buf ^= 1;
  }

  // C/D layout: VGPR r -> row (r + g*8), col = ml
#pragma unroll
  for (int r = 0; r < 8; ++r) {
    O[(bm + wm * 16 + r + g * 8) * (long)N + bn + wn * 16 + ml] = acc[r];
  }
}

// ---------------------------------------------------------------------------
// Kernel 2: causal flash attention, head_dim = 4.
// One wave handles one (b, h, 16-row Q tile). Q*K^T = ONE wmma_f32_16x16x4
// per 16-key tile. Online softmax in C-layout via __shfl_xor (16-lane row
// groups). P and the 16x4 V tile staged in per-wave LDS (LDS is in-order per
// wave -> no block barrier needed in the wave-divergent causal loop).
// ---------------------------------------------------------------------------
__global__ __launch_bounds__(256) void attn_flash(const float* __restrict__ qkv,
                                                  float* __restrict__ y) {
  __shared__ float Pbuf[8][16 * 17];  // score/prob tile, pad-17 rows
  __shared__ float Vbuf[8][16 * 4];   // V tile
  __shared__ float Abuf[8][16];       // per-row rescale alpha
  __shared__ float Lbuf[8][16];       // per-row softmax denom

  const int t    = threadIdx.x;
  const int lane = t & 31;
  const int w    = t >> 5;
  const int ml   = lane & 15;
  const int g    = lane >> 4;

  const int wg = blockIdx.x * 8 + w;          // 0 .. 16383
  const int qt = wg & 63;                      // Q tile (T/16 = 64)
  const int h  = (wg >> 6) & (NH - 1);         // head
  const int bi = wg >> 13;                     // batch

  float* P  = Pbuf[w];
  float* V  = Vbuf[w];
  float* Al = Abuf[w];
  float* Ll = Lbuf[w];

  const long base = (long)bi * Tt * (3 * Cc);

  // Q fragment (loaded once): A-matrix 16x4, lane ml = row, K pair = g*2
  const v2f aq =
      *(const v2f*)(qkv + base + (long)(qt * 16 + ml) * (3 * Cc) + h * 4 + g * 2);

  float m[8], l[8];
#pragma unroll
  for (int r = 0; r < 8; ++r) { m[r] = -1e30f; l[r] = 0.0f; }
  float o0 = 0.f, o1 = 0.f, o2 = 0.f, o3 = 0.f;

  const v8f zero = {};

  for (int j = 0; j <= qt; ++j) {
    // K^T fragment: B-matrix 4x16, lane ml = key col, K pair = g*2
    v2f bkf = *(const v2f*)(qkv + base + (long)(j * 16 + ml) * (3 * Cc) +
                            Cc + h * 4 + g * 2);
    v8f s = wmma_f32_16x16x4(aq, bkf, zero);

    // Stage V tile (16 rows x 4) into LDS
    if (lane < 16) {
      *(float4*)(&V[lane * 4]) = *(const float4*)(
          qkv + base + (long)(j * 16 + lane) * (3 * Cc) + 2 * Cc + h * 4);
    }

    const bool diag = (j == qt);
#pragma unroll
    for (int r = 0; r < 8; ++r) {
      const int row = r + g * 8;               // row within tile
      float sv = s[r] * 0.25f;                 // scale = 1/HD
      if (diag && (ml > row)) sv = -1e30f;     // causal mask
      // row max across the 16 lanes owning this row
      float rm = sv;
      rm = fmaxf(rm, __shfl_xor(rm, 1));
      rm = fmaxf(rm, __shfl_xor(rm, 2));
      rm = fmaxf(rm, __shfl_xor(rm, 4));
      rm = fmaxf(rm, __shfl_xor(rm, 8));
      const float mn = fmaxf(m[r], rm);
      const float p  = __expf(sv - mn);
      float rs = p;
      rs += __shfl_xor(rs, 1);
      rs += __shfl_xor(rs, 2);
      rs += __shfl_xor(rs, 4);
      rs += __shfl_xor(rs, 8);
      const float alpha = __expf(m[r] - mn);
      l[r] = l[r] * alpha + rs;
      m[r] = mn;
      P[row * 17 + ml] = p;
      if (ml == 0) Al[row] = alpha;
    }
    asm volatile("" ::: "memory");  // keep LDS store->load order (HW in-order)

    // O update: lane (ml) owns output row ml (lanes 16-31 shadow-compute)
    const float am = Al[ml];
    float a0 = 0.f, a1 = 0.f, a2 = 0.f, a3 = 0.f;
#pragma unroll
    for (int k = 0; k < 16; ++k) {
      const float pk = P[ml * 17 + k];
      a0 += pk * V[k * 4 + 0];
      a1 += pk * V[k * 4 + 1];
      a2 += pk * V[k * 4 + 2];
      a3 += pk * V[k * 4 + 3];
    }
    o0 = o0 * am + a0;
    o1 = o1 * am + a1;
    o2 = o2 * am + a2;
    o3 = o3 * am + a3;
    asm volatile("" ::: "memory");  // WAR on P/V/Al for next iteration
  }

#pragma unroll
  for (int r = 0; r < 8; ++r)
    if (ml == 0) Ll[r + g * 8] = l[r];
  asm volatile("" ::: "memory");

  if (lane < 16) {
    const float inv = 1.0f / Ll[lane];
    const long qr = (long)qt * 16 + lane;
    *(float4*)(y + ((long)bi * Tt + qr) * Cc + h * 4) =
        make_float4(o0 * inv, o1 * inv, o2 * inv, o3 * inv);
  }
}

// ---------------------------------------------------------------------------
// Launch: qkv = x @ w_attn^T  ->  flash attention  ->  out = y @ w_proj^T
// ---------------------------------------------------------------------------
extern "C" void kernel_launch(void* const* d_in, const int* in_sizes, int n_in,
                              void* d_out, int out_size, void* d_ws,
                              size_t ws_size, hipStream_t stream) {
  (void)in_sizes; (void)n_in; (void)out_size; (void)ws_size;
  const float* x      = (const float*)d_in[0];
  const float* w_attn = (const float*)d_in[1];
  const float* w_proj = (const float*)d_in[2];
  float* out = (float*)d_out;

  float* qkv = (float*)d_ws;                          // (B*T) x 3C = 12.6 MB
  float* yws = qkv + (size_t)Bb * Tt * 3 * Cc;        // (B*T) x C  =  4.2 MB

  const int M = Bb * Tt;  // 2048
  dim3 blk(256);

  // QKV projection: M=2048, N=1536, K=512
  gemm_xwt<<<dim3((3 * Cc) / 64, M / 32), blk, 0, stream>>>(
      x, w_attn, qkv, M, 3 * Cc, Cc);

  // Attention: B*NH*(T/16) = 16384 waves, 8 waves/block
  attn_flash<<<dim3((Bb * NH * (Tt / 16)) / 8), blk, 0, stream>>>(qkv, yws);

  // Output projection: M=2048, N=512, K=512
  gemm_xwt<<<dim3(Cc / 64, M / 32), blk, 0, stream>>>(
      yws, w_proj, out, M, Cc, Cc);
}